// GatClassifier_40235253628946
// MI455X (gfx1250) — compile-verified
//
#include <hip/hip_runtime.h>
#include <hip/hip_bf16.h>
#include <math.h>

typedef __attribute__((ext_vector_type(2))) float v2f;
typedef __attribute__((ext_vector_type(8))) float v8f;

#define FIN   128   // feature width for both layers (IN_CH == HEADS*HID == 128)
#define HEADS 4
#define HID   32
#define NEG_SLOPE 0.2f

// ---------------------------------------------------------------------------
// float atomic max via signed/unsigned integer ordering trick
// (positive floats ordered as ints; negative floats reverse-ordered as uints)
// ---------------------------------------------------------------------------
__device__ __forceinline__ void atomicMaxF(float* addr, float v) {
  if (v >= 0.0f)
    atomicMax((int*)addr, __float_as_int(v));
  else
    atomicMin((unsigned int*)addr, __float_as_uint(v));
}

__global__ void fill_f32_kernel(float* __restrict__ p, float v, long long n) {
  long long i = (long long)blockIdx.x * blockDim.x + threadIdx.x;
  if (i < n) p[i] = v;
}

// ---------------------------------------------------------------------------
// C[n x 128] = A[n x 128] @ W[128 x 128] using V_WMMA_F32_16X16X4_F32.
// Block = 256 threads = 8 waves; wave w computes the 16x16 tile at columns
// [16w, 16w+16) for the 16-row stripe blockIdx.x. n is a multiple of 16.
//
// A operand layout (ISA 7.12.2, 32-bit A 16x4): lanes 0-15 hold M=lane with
// K = k+0 (a.x), k+1 (a.y); lanes 16-31 hold M=lane-16 with K = k+2, k+3.
// B operand mirrors it with N = lane%16. C/D: VGPR r holds row r (lanes 0-15)
// / row r+8 (lanes 16-31), column = lane%16.
// ---------------------------------------------------------------------------
__global__ __launch_bounds__(256) void gemm128_wmma(const float* __restrict__ A,
                                                    const float* __restrict__ W,
                                                    float* __restrict__ C) {
  const int wave  = threadIdx.x >> 5;
  const int lane  = threadIdx.x & 31;
  const int lhalf = lane >> 4;      // 0 or 1
  const int l15   = lane & 15;
  const int row0  = blockIdx.x << 4;
  const int col0  = wave << 4;

  const float* Arow = A + (size_t)(row0 + l15) * FIN + 2 * lhalf;
  const float* Wcol = W + (size_t)(2 * lhalf) * FIN + col0 + l15;

  v8f acc = {};
#pragma unroll
  for (int k = 0; k < FIN; k += 4) {
    v2f a, b;
    a.x = Arow[k + 0];
    a.y = Arow[k + 1];
    b.x = Wcol[(size_t)k * FIN];
    b.y = Wcol[(size_t)(k + 1) * FIN];
    acc = __builtin_amdgcn_wmma_f32_16x16x4_f32(
        /*neg_a=*/false, a, /*neg_b=*/false, b,
        /*c_mod=*/(short)0, acc, /*reuse_a=*/false, /*reuse_b=*/false);
  }
#pragma unroll
  for (int r = 0; r < 8; ++r)
    C[(size_t)(row0 + lhalf * 8 + r) * FIN + col0 + l15] = acc[r];
}

// ---------------------------------------------------------------------------
// als[n,h] = <h[n,h,:], a_src[h,:]> ; ald likewise. One thread per (node,head).
// ---------------------------------------------------------------------------
__global__ void attn_logits_kernel(const float* __restrict__ h,
                                   const float* __restrict__ a_src,
                                   const float* __restrict__ a_dst,
                                   float* __restrict__ als,
                                   float* __restrict__ ald, int n) {
  int i = blockIdx.x * blockDim.x + threadIdx.x;
  if (i >= n * HEADS) return;
  int head = i & (HEADS - 1);
  int node = i >> 2;
  const float* hp = h + (size_t)node * FIN + head * HID;
  const float* as = a_src + head * HID;
  const float* ad = a_dst + head * HID;
  float s = 0.f, d = 0.f;
#pragma unroll
  for (int c = 0; c < HID; ++c) {
    float v = hp[c];
    s = fmaf(v, as[c], s);
    d = fmaf(v, ad[c], d);
  }
  als[i] = s;
  ald[i] = d;
}

// ---------------------------------------------------------------------------
// Edge pass 1: e = leaky_relu(als[src]+ald[dst]); segment max into m[dst].
// Self-loop edges are appended virtually: eid >= nE -> src = dst = eid - nE.
// ---------------------------------------------------------------------------
__global__ void edge_max_kernel(const int* __restrict__ srcA,
                                const int* __restrict__ dstA,
                                long long nE, int nN,
                                const float* __restrict__ als,
                                const float* __restrict__ ald,
                                float* __restrict__ ebuf,
                                float* __restrict__ m) {
  long long i = (long long)blockIdx.x * blockDim.x + threadIdx.x;
  long long tot = (nE + nN) * HEADS;
  if (i >= tot) return;
  int head = (int)(i & (HEADS - 1));
  long long eid = i >> 2;
  int s, d;
  if (eid < nE) { s = srcA[eid]; d = dstA[eid]; }
  else          { s = d = (int)(eid - nE); }
  float v = als[(size_t)s * HEADS + head] + ald[(size_t)d * HEADS + head];
  v = (v > 0.f) ? v : NEG_SLOPE * v;
  ebuf[i] = v;
  atomicMaxF(&m[(size_t)d * HEADS + head], v);
}

// ---------------------------------------------------------------------------
// Edge pass 2: ex = exp(e - m[dst]); segment sum into den[dst]; ebuf <- ex.
// ---------------------------------------------------------------------------
__global__ void edge_expsum_kernel(const int* __restrict__ dstA,
                                   long long nE, int nN,
                                   const float* __restrict__ m,
                                   float* __restrict__ ebuf,
                                   float* __restrict__ den) {
  long long i = (long long)blockIdx.x * blockDim.x + threadIdx.x;
  long long tot = (nE + nN) * HEADS;
  if (i >= tot) return;
  int head = (int)(i & (HEADS - 1));
  long long eid = i >> 2;
  int d = (eid < nE) ? dstA[eid] : (int)(eid - nE);
  float ex = __expf(ebuf[i] - m[(size_t)d * HEADS + head]);
  ebuf[i] = ex;
  atomicAdd(&den[(size_t)d * HEADS + head], ex);
}

// ---------------------------------------------------------------------------
// Edge pass 3: out[dst, c] += (ex / den[dst]) * h[src, c].
// One thread per (edge, channel): channel-coalesced gather + float atomic add.
// ---------------------------------------------------------------------------
__global__ void edge_aggregate_kernel(const int* __restrict__ srcA,
                                      const int* __restrict__ dstA,
                                      long long nE, int nN,
                                      const float* __restrict__ ebuf,
                                      const float* __restrict__ den,
                                      const float* __restrict__ h,
                                      float* __restrict__ out) {
  long long i = (long long)blockIdx.x * blockDim.x + threadIdx.x;
  long long tot = (nE + nN) * (long long)FIN;
  if (i >= tot) return;
  int c = (int)(i & (FIN - 1));
  long long eid = i >> 7;
  int head = c >> 5;  // c / HID
  int s, d;
  if (eid < nE) { s = srcA[eid]; d = dstA[eid]; }
  else          { s = d = (int)(eid - nE); }
  float attn = ebuf[eid * HEADS + head] / den[(size_t)d * HEADS + head];
  atomicAdd(&out[(size_t)d * FIN + c], attn * h[(size_t)s * FIN + c]);
}

// ---------------------------------------------------------------------------
// out = elu(out + bias)  (alpha = 1)
// ---------------------------------------------------------------------------
__global__ void bias_elu_kernel(float* __restrict__ out,
                                const float* __restrict__ bias, long long n128) {
  long long i = (long long)blockIdx.x * blockDim.x + threadIdx.x;
  if (i >= n128) return;
  float v = out[i] + bias[i & (FIN - 1)];
  out[i] = (v > 0.f) ? v : (__expf(v) - 1.f);
}

// ---------------------------------------------------------------------------
// Global mean-pool (segment sum + counts via atomics).
// ---------------------------------------------------------------------------
__global__ void pool_kernel(const float* __restrict__ act,
                            const int* __restrict__ batch,
                            float* __restrict__ sums,
                            float* __restrict__ cnts, int n) {
  long long i = (long long)blockIdx.x * blockDim.x + threadIdx.x;
  long long tot = (long long)n * FIN;
  if (i >= tot) return;
  int c = (int)(i & (FIN - 1));
  int node = (int)(i >> 7);
  int g = batch[node];
  atomicAdd(&sums[(size_t)g * FIN + c], act[i]);
  if (c == 0) atomicAdd(&cnts[g], 1.0f);
}

__global__ void readout_kernel(const float* __restrict__ sums,
                               const float* __restrict__ cnts,
                               const float* __restrict__ lin_w,
                               const float* __restrict__ lin_b,
                               float* __restrict__ out, int G) {
  int g = blockIdx.x * blockDim.x + threadIdx.x;
  if (g >= G) return;
  float cnt = cnts[g];
  cnt = (cnt > 1.0f) ? cnt : 1.0f;
  float acc = 0.f;
#pragma unroll 8
  for (int c = 0; c < FIN; ++c)
    acc = fmaf(sums[(size_t)g * FIN + c] / cnt, lin_w[c], acc);
  out[g] = acc + lin_b[0];
}

// ---------------------------------------------------------------------------
// Host launcher
// ---------------------------------------------------------------------------
static inline unsigned int gridFor(long long n, int block) {
  return (unsigned int)((n + block - 1) / block);
}

extern "C" void kernel_launch(void* const* d_in, const int* in_sizes, int n_in,
                              void* d_out, int out_size, void* d_ws, size_t ws_size,
                              hipStream_t stream) {
  const float* x    = (const float*)d_in[0];
  const int*   eidx = (const int*)d_in[1];
  const int*   batch= (const int*)d_in[2];
  const float* W1   = (const float*)d_in[3];
  const float* a1s  = (const float*)d_in[4];
  const float* a1d  = (const float*)d_in[5];
  const float* b1   = (const float*)d_in[6];
  const float* W2   = (const float*)d_in[7];
  const float* a2s  = (const float*)d_in[8];
  const float* a2d  = (const float*)d_in[9];
  const float* b2   = (const float*)d_in[10];
  const float* linw = (const float*)d_in[11];
  const float* linb = (const float*)d_in[12];

  const int       N  = in_sizes[0] / FIN;           // 50000 (multiple of 16)
  const long long E  = in_sizes[1] / 2;             // 1,600,000
  const int       G  = out_size;                    // 64 graphs
  const long long Ep = E + N;                       // with self loops
  const int* srcA = eidx;
  const int* dstA = eidx + E;

  // ---- workspace carve-out (256B aligned) ----
  char* wsp = (char*)d_ws;
  auto carve = [&](size_t bytes) -> float* {
    float* p = (float*)wsp;
    wsp += (bytes + 255) & ~(size_t)255;
    return p;
  };
  float* bufA   = carve((size_t)N * FIN * 4);       // act buffer / layer output
  float* bufB   = carve((size_t)N * FIN * 4);       // h = input @ W
  float* als    = carve((size_t)N * HEADS * 4);
  float* ald    = carve((size_t)N * HEADS * 4);
  float* mbuf   = carve((size_t)N * HEADS * 4);
  float* den    = carve((size_t)N * HEADS * 4);
  float* ebuf   = carve((size_t)Ep * HEADS * 4);
  float* pooled = carve((size_t)G * FIN * 4);
  float* cnts   = carve((size_t)G * 4);

  const int BLK = 256;
  const long long nNH   = (long long)N * HEADS;
  const long long nN128 = (long long)N * FIN;
  const long long nEpH  = Ep * HEADS;
  const long long nEp128= Ep * (long long)FIN;

  // ---- one GAT layer: in -> bufB (=in@W) -> softmax-aggregate -> outAct ----
  auto gat_layer = [&](const float* in, const float* W, const float* as,
                       const float* ad, const float* bias, float* outAct) {
    gemm128_wmma<<<N / 16, 256, 0, stream>>>(in, W, bufB);
    attn_logits_kernel<<<gridFor(nNH, BLK), BLK, 0, stream>>>(bufB, as, ad, als, ald, N);
    fill_f32_kernel<<<gridFor(nNH, BLK), BLK, 0, stream>>>(mbuf, -INFINITY, nNH);
    fill_f32_kernel<<<gridFor(nNH, BLK), BLK, 0, stream>>>(den, 0.0f, nNH);
    fill_f32_kernel<<<gridFor(nN128, BLK), BLK, 0, stream>>>(outAct, 0.0f, nN128);
    edge_max_kernel<<<gridFor(nEpH, BLK), BLK, 0, stream>>>(srcA, dstA, E, N, als, ald, ebuf, mbuf);
    edge_expsum_kernel<<<gridFor(nEpH, BLK), BLK, 0, stream>>>(dstA, E, N, mbuf, ebuf, den);
    edge_aggregate_kernel<<<gridFor(nEp128, BLK), BLK, 0, stream>>>(srcA, dstA, E, N, ebuf, den, bufB, outAct);
    bias_elu_kernel<<<gridFor(nN128, BLK), BLK, 0, stream>>>(outAct, bias, nN128);
  };

  gat_layer(x,    W1, a1s, a1d, b1, bufA);   // layer 1: x -> act1 (bufA)
  gat_layer(bufA, W2, a2s, a2d, b2, bufA);   // layer 2: act1 -> act2 (bufA; bufA
                                             // only read by the GEMM, which
                                             // completes before it is zeroed)

  fill_f32_kernel<<<gridFor((long long)G * FIN, BLK), BLK, 0, stream>>>(pooled, 0.0f, (long long)G * FIN);
  fill_f32_kernel<<<1, BLK, 0, stream>>>(cnts, 0.0f, G);
  pool_kernel<<<gridFor(nN128, BLK), BLK, 0, stream>>>(bufA, batch, pooled, cnts, N);
  readout_kernel<<<1, 64, 0, stream>>>(pooled, cnts, linw, linb, (float*)d_out, G);
}